// DDFModule_24927990186533
// MI455X (gfx1250) — compile-verified
//
#include <hip/hip_runtime.h>
#include <math.h>

typedef __attribute__((ext_vector_type(2))) float v2f;
typedef __attribute__((ext_vector_type(8))) float v8f;

#define AS1 __attribute__((address_space(1)))
#define AS3 __attribute__((address_space(3)))

#define BB 8
#define CC 256
#define HH 64
#define WW 64
#define TS 8            // spatial tile 8x8
#define HALO 10         // TS + 2
#define HH2 (HALO*HALO) // 100
#define NPIX (TS*TS)    // 64 pixels per tile
#define NPIX_TOT (BB*HH*WW)

#if defined(__HIP_DEVICE_COMPILE__) && __has_builtin(__builtin_amdgcn_global_load_async_to_lds_b32)
#define HAS_ASYNC_LDS 1
#else
#define HAS_ASYNC_LDS 0
#endif

// ---------------------------------------------------------------------------
// Kernel 1: global average pool over HxW per (b,c)
// ---------------------------------------------------------------------------
__global__ __launch_bounds__(256) void pool_kernel(const float* __restrict__ x,
                                                   float* __restrict__ pooled) {
    const int bc = blockIdx.x;              // b*C + c
    __shared__ float red[256];
    const float* px = x + (size_t)bc * (HH * WW);
    float s = 0.f;
    for (int i = threadIdx.x; i < HH * WW; i += 256) s += px[i];
    red[threadIdx.x] = s;
    __syncthreads();
    for (int st = 128; st > 0; st >>= 1) {
        if (threadIdx.x < st) red[threadIdx.x] += red[threadIdx.x + st];
        __syncthreads();
    }
    if (threadIdx.x == 0) pooled[bc] = red[0] * (1.f / (HH * WW));
}

// ---------------------------------------------------------------------------
// Kernel 2: ECA conv1d (k=3, zero pad) + sigmoid; also zero BN accumulators
// ---------------------------------------------------------------------------
__global__ __launch_bounds__(256) void attn_kernel(const float* __restrict__ pooled,
                                                   const float* __restrict__ weca,
                                                   float* __restrict__ attn,
                                                   float* __restrict__ gsum,
                                                   float* __restrict__ gsumsq) {
    const int b = blockIdx.x, c = threadIdx.x;
    const float pm = pooled[b * CC + c];
    const float pl = (c > 0)      ? pooled[b * CC + c - 1] : 0.f;
    const float pr = (c < CC - 1) ? pooled[b * CC + c + 1] : 0.f;
    const float e = weca[0] * pl + weca[1] * pm + weca[2] * pr;
    attn[b * CC + c] = 1.f / (1.f + expf(-e));
    if (b == 0) { gsum[c] = 0.f; gsumsq[c] = 0.f; }
}

// ---------------------------------------------------------------------------
// Kernel 3: fused dynamic-filter + channel-attn + projection, WMMA f32 16x16x4
//   grid: 512 WGs (one per (b, 8x8 tile)), 512 threads = 16 waves.
//   Each wave owns one M-tile (16 channels) and all 4 N-tiles (4x v8f acc)
//   so each A fragment feeds 4 back-to-back WMMAs.
// ---------------------------------------------------------------------------
__global__ __launch_bounds__(512) void ddf_main_kernel(
    const float* __restrict__ x,
    const float* __restrict__ Wf,    // (C*9, C) row-major
    const float* __restrict__ bf,    // (C*9)
    const float* __restrict__ Wp,    // (C, C) row-major
    const float* __restrict__ bp,    // (C)
    const float* __restrict__ attn,  // (B, C)
    float* __restrict__ y,           // (B, C, H, W) workspace
    float* __restrict__ gsum, float* __restrict__ gsumsq) {

    extern __shared__ float smem[];
    float* lds_x = smem;                       // 256*100 floats (halo tile)
    float* lds_f = smem + CC * HH2;            // 256*64 floats (fused buffer)
    float* lds_s = lds_f + CC * NPIX;          // 256 (BN sum partials)
    float* lds_q = lds_s + CC;                 // 256 (BN sumsq partials)
    float* lds_a = lds_q + CC;                 // 256 (attn per channel)

    const int wg  = blockIdx.x;                // 0..511
    const int b   = wg >> 6;                   // 64 tiles per image
    const int til = wg & 63;
    const int y0 = (til >> 3) * TS, x0 = (til & 7) * TS;

    const int t = threadIdx.x;

    // ---- load halo tile (zero-padded) via async global->LDS DMA ----
    for (int i = t; i < CC * HH2; i += 512) {
        const int c = i / HH2, q = i % HH2;
        const int hy = y0 + (q / HALO) - 1, hx = x0 + (q % HALO) - 1;
        const bool in = (hy >= 0 && hy < HH && hx >= 0 && hx < WW);
#if HAS_ASYNC_LDS
        if (in) {
            const float* g = x + (((size_t)b * CC + c) * HH + hy) * WW + hx;
            __builtin_amdgcn_global_load_async_to_lds_b32(
                (AS1 int*)g, (AS3 int*)(lds_x + i), 0, 0);
        } else {
            lds_x[i] = 0.f;
        }
#else
        lds_x[i] = in ? x[(((size_t)b * CC + c) * HH + hy) * WW + hx] : 0.f;
#endif
    }
    if (t < CC) {
        lds_s[t] = 0.f; lds_q[t] = 0.f;
        lds_a[t] = attn[b * CC + t];
    }
#if HAS_ASYNC_LDS
#if __has_builtin(__builtin_amdgcn_s_wait_asynccnt)
    __builtin_amdgcn_s_wait_asynccnt(0);
#else
    asm volatile("s_wait_asynccnt 0" ::: "memory");
#endif
#endif
    __syncthreads();

    const int lane = t & 31;
    const int wave = t >> 5;       // 0..15 == M-tile index
    const int hi = lane >> 4;      // 0/1: selects K pair {2h,2h+1} in A/B frags
    const int lo = lane & 15;
    const int mt = wave;

    int pcol[4], cen[4];
    #pragma unroll
    for (int nt = 0; nt < 4; ++nt) {
        const int p = nt * 16 + lo;
        pcol[nt] = p;
        cen[nt] = ((p >> 3) + 1) * HALO + ((p & 7) + 1);
    }

    // ================= Stage 1: dynamic filter + spatial feature ===========
    // Per tap k: filt_k (256 out-ch x 64 px) = Wk(256x256) @ xtile(256x64)
    // spatial[c,p] = sum_k patch[c,k,p] * filt_k[c,p]
    v8f spat[4];
    #pragma unroll
    for (int nt = 0; nt < 4; ++nt)
        #pragma unroll
        for (int j = 0; j < 8; ++j) spat[nt][j] = 0.f;

    const int rowc = mt * 16 + lo;             // A-frag row (out channel)
    for (int k = 0; k < 9; ++k) {
        v8f acc[4];
        #pragma unroll
        for (int j = 0; j < 8; ++j) {          // C init = b_filter (same per nt)
            const float bfv = bf[(mt * 16 + j + 8 * hi) * 9 + k];
            acc[0][j] = bfv; acc[1][j] = bfv; acc[2][j] = bfv; acc[3][j] = bfv;
        }
        const float* __restrict__ arow = Wf + ((size_t)(rowc * 9 + k)) * CC;
        for (int kk = 0; kk < CC / 4; ++kk) {
            const int kb = kk * 4 + 2 * hi;
            v2f a;
            a[0] = arow[kb];
            a[1] = arow[kb + 1];
            #pragma unroll
            for (int nt = 0; nt < 4; ++nt) {
                v2f bv;
                bv[0] = lds_x[kb * HH2 + cen[nt]];
                bv[1] = lds_x[(kb + 1) * HH2 + cen[nt]];
                acc[nt] = __builtin_amdgcn_wmma_f32_16x16x4_f32(
                    false, a, false, bv, (short)0, acc[nt], false, false);
            }
        }
        const int dy = k / 3, dx = k % 3;
        #pragma unroll
        for (int nt = 0; nt < 4; ++nt) {
            const int off = ((pcol[nt] >> 3) + dy) * HALO + ((pcol[nt] & 7) + dx);
            #pragma unroll
            for (int j = 0; j < 8; ++j) {
                const int c = mt * 16 + j + 8 * hi;
                spat[nt][j] += lds_x[c * HH2 + off] * acc[nt][j];
            }
        }
    }
    // fuse: 0.5 * attn*x + 0.5 * spatial -> LDS fused buffer
    #pragma unroll
    for (int nt = 0; nt < 4; ++nt) {
        #pragma unroll
        for (int j = 0; j < 8; ++j) {
            const int c = mt * 16 + j + 8 * hi;
            const float xv = lds_x[c * HH2 + cen[nt]];
            lds_f[c * NPIX + pcol[nt]] = 0.5f * (lds_a[c] * xv) + 0.5f * spat[nt][j];
        }
    }
    __syncthreads();

    // ================= Stage 2: 1x1 projection GEMM + BN stats =============
    // y(256 x 64px) = W_proj(256x256) @ fused(256x64) + b_proj
    {
        v8f acc[4];
        #pragma unroll
        for (int j = 0; j < 8; ++j) {
            const float bpv = bp[mt * 16 + j + 8 * hi];
            acc[0][j] = bpv; acc[1][j] = bpv; acc[2][j] = bpv; acc[3][j] = bpv;
        }
        const float* __restrict__ arow = Wp + (size_t)rowc * CC;
        for (int kk = 0; kk < CC / 4; ++kk) {
            const int kb = kk * 4 + 2 * hi;
            v2f a;
            a[0] = arow[kb];
            a[1] = arow[kb + 1];
            #pragma unroll
            for (int nt = 0; nt < 4; ++nt) {
                v2f bv;
                bv[0] = lds_f[kb * NPIX + pcol[nt]];
                bv[1] = lds_f[(kb + 1) * NPIX + pcol[nt]];
                acc[nt] = __builtin_amdgcn_wmma_f32_16x16x4_f32(
                    false, a, false, bv, (short)0, acc[nt], false, false);
            }
        }
        #pragma unroll
        for (int nt = 0; nt < 4; ++nt) {
            const int gy = y0 + (pcol[nt] >> 3), gx = x0 + (pcol[nt] & 7);
            #pragma unroll
            for (int j = 0; j < 8; ++j) {
                const int o = mt * 16 + j + 8 * hi;
                const float v = acc[nt][j];
                y[(((size_t)b * CC + o) * HH + gy) * WW + gx] = v;
                atomicAdd(&lds_s[o], v);
                atomicAdd(&lds_q[o], v * v);
            }
        }
    }
    __syncthreads();
    if (t < CC) {
        atomicAdd(&gsum[t], lds_s[t]);
        atomicAdd(&gsumsq[t], lds_q[t]);
    }
}

// ---------------------------------------------------------------------------
// Kernel 4: finalize BN scale/shift per channel
// ---------------------------------------------------------------------------
__global__ __launch_bounds__(256) void bnstat_kernel(
    const float* __restrict__ gsum, const float* __restrict__ gsumsq,
    const float* __restrict__ gamma, const float* __restrict__ beta,
    float* __restrict__ scale, float* __restrict__ shift) {
    const int c = threadIdx.x;
    const float n = (float)NPIX_TOT;
    const float m = gsum[c] / n;
    const float var = gsumsq[c] / n - m * m;
    const float sc = gamma[c] * rsqrtf(var + 1e-5f);
    scale[c] = sc;
    shift[c] = beta[c] - m * sc;
}

// ---------------------------------------------------------------------------
// Kernel 5: apply BN (vectorized float4)
// ---------------------------------------------------------------------------
__global__ __launch_bounds__(256) void bnapply_kernel(
    const float* __restrict__ y, const float* __restrict__ scale,
    const float* __restrict__ shift, float* __restrict__ out) {
    const int n4 = (BB * CC * HH * WW) / 4;
    const int i = blockIdx.x * blockDim.x + threadIdx.x;
    if (i >= n4) return;
    const float4 v = ((const float4*)y)[i];
    const int c = (i >> 10) & (CC - 1);   // (i*4)/(H*W) % C, H*W=4096
    const float sc = scale[c], sh = shift[c];
    float4 o;
    o.x = v.x * sc + sh; o.y = v.y * sc + sh;
    o.z = v.z * sc + sh; o.w = v.w * sc + sh;
    ((float4*)out)[i] = o;
}

// ---------------------------------------------------------------------------
extern "C" void kernel_launch(void* const* d_in, const int* in_sizes, int n_in,
                              void* d_out, int out_size, void* d_ws, size_t ws_size,
                              hipStream_t stream) {
    const float* x     = (const float*)d_in[0];
    const float* Wf    = (const float*)d_in[1];
    const float* bf    = (const float*)d_in[2];
    const float* weca  = (const float*)d_in[3];
    const float* Wp    = (const float*)d_in[4];
    const float* bp    = (const float*)d_in[5];
    const float* gamma = (const float*)d_in[6];
    const float* beta  = (const float*)d_in[7];

    float* ws     = (float*)d_ws;
    float* y      = ws;                          // B*C*H*W
    float* pooled = y + (size_t)BB * CC * HH * WW;
    float* attn   = pooled + BB * CC;
    float* gsum   = attn + BB * CC;
    float* gsumsq = gsum + CC;
    float* scale  = gsumsq + CC;
    float* shift  = scale + CC;

    pool_kernel<<<BB * CC, 256, 0, stream>>>(x, pooled);
    attn_kernel<<<BB, 256, 0, stream>>>(pooled, weca, attn, gsum, gsumsq);

    const size_t smem = (size_t)(CC * HH2 + CC * NPIX + 3 * CC) * sizeof(float);
    ddf_main_kernel<<<BB * (HH / TS) * (WW / TS), 512, smem, stream>>>(
        x, Wf, bf, Wp, bp, attn, y, gsum, gsumsq);

    bnstat_kernel<<<1, 256, 0, stream>>>(gsum, gsumsq, gamma, beta, scale, shift);

    const int n4 = (BB * CC * HH * WW) / 4;
    bnapply_kernel<<<(n4 + 255) / 256, 256, 0, stream>>>(y, scale, shift, (float*)d_out);
}